// cklXLoss_4604204941999
// MI455X (gfx1250) — compile-verified
//
#include <hip/hip_runtime.h>
#include <math.h>

typedef __attribute__((ext_vector_type(2))) float v2f;
typedef __attribute__((ext_vector_type(8))) float v8f;

#define DIMS 128
#define MU_F 1.0f
#define EPS_F 1e-8f

// ---------------------------------------------------------------------------
// Kernel 1: squared row norms. One wave32 per row: 32 lanes x float4 = 128.
// ---------------------------------------------------------------------------
__global__ void sqnorm_kernel(const float* __restrict__ x,
                              float* __restrict__ sq, int nrows) {
  int wave = (blockIdx.x * blockDim.x + threadIdx.x) >> 5;
  int lane = threadIdx.x & 31;
  if (wave >= nrows) return;
  const float4* row = (const float4*)(x + (size_t)wave * DIMS);
  float4 v = row[lane];
  float s = v.x * v.x + v.y * v.y + v.z * v.z + v.w * v.w;
#pragma unroll
  for (int off = 16; off > 0; off >>= 1) s += __shfl_xor(s, off, 32);
  if (lane == 0) sq[wave] = s;
}

// ---------------------------------------------------------------------------
// Kernel 2: one wave32 handles 16 triplets via V_WMMA_F32_16X16X4_F32.
//   A (16x4)  = anchor rows,  B (4x16) = pos/neg rows as columns.
//   32 K-steps of 4 cover D=128. Diagonal of the 16x16 accumulator gives the
//   per-triplet dot products in full f32.
// Fragment addressing (identical for A and B per ISA VGPR layouts):
//   lane L supplies {x[r][k], x[r][k+1]}, r = idx[L&15], k = kk*4 + (L<16?0:2)
// ---------------------------------------------------------------------------
__global__ void triplet_wmma_kernel(const float* __restrict__ x,
                                    const int* __restrict__ trip,
                                    const float* __restrict__ sq,
                                    float* __restrict__ block_partials,
                                    int T) {
  __shared__ float wsum[8];
  const int lane = threadIdx.x & 31;
  const int waveInBlock = threadIdx.x >> 5;
  const int wave = blockIdx.x * (blockDim.x >> 5) + waveInBlock;
  const int base = wave * 16;

  float contrib = 0.0f;

  if (base < T) {
    const int m = lane & 15;
    int ti = base + m;
    if (ti > T - 1) ti = T - 1;              // clamp keeps EXEC all-1s for WMMA
    const int a_idx = trip[ti * 3 + 0];
    const int p_idx = trip[ti * 3 + 1];
    const int n_idx = trip[ti * 3 + 2];

    const float* __restrict__ xa = x + (size_t)a_idx * DIMS;
    const float* __restrict__ xp = x + (size_t)p_idx * DIMS;
    const float* __restrict__ xn = x + (size_t)n_idx * DIMS;
    const int koff = (lane < 16) ? 0 : 2;

    v8f c_an = {};
    v8f c_ap = {};
#pragma unroll
    for (int kk = 0; kk < 32; ++kk) {
      const int k = kk * 4 + koff;
      v2f av = *(const v2f*)(xa + k);
      v2f nv = *(const v2f*)(xn + k);
      v2f pv = *(const v2f*)(xp + k);
      // 8 args: (neg_a, A, neg_b, B, c_mod, C, reuse_a, reuse_b)
      c_an = __builtin_amdgcn_wmma_f32_16x16x4_f32(
          false, av, false, nv, (short)0, c_an, false, false);
      c_ap = __builtin_amdgcn_wmma_f32_16x16x4_f32(
          false, av, false, pv, (short)0, c_ap, false, false);
    }

    // Diagonal extraction: element (M=v,N=v) lives in c[v] on lane v;
    // (M=v+8,N=v+8) lives in c[v] on lane v+24. Each holder computes its
    // triplet's (negated) log term; other lanes contribute 0.
#pragma unroll
    for (int v = 0; v < 8; ++v) {
      const bool holds = (lane == v) || (lane == v + 24);
      if (holds && (base + m) < T) {
        const float dot_an = c_an[v];
        const float dot_ap = c_ap[v];
        const float sqa = sq[a_idx];
        const float d_an = sqa + sq[n_idx] - 2.0f * dot_an;
        const float d_ap = sqa + sq[p_idx] - 2.0f * dot_ap;
        const float numer = fmaxf(MU_F + d_an, EPS_F);
        const float denom = fmaxf(2.0f * MU_F + d_ap + d_an, EPS_F);
        contrib += logf(denom) - logf(numer);   // pre-negated
      }
    }
  }

  // Wave reduce (wave32 butterfly), then deterministic block reduce.
#pragma unroll
  for (int off = 16; off > 0; off >>= 1) contrib += __shfl_xor(contrib, off, 32);
  if (lane == 0) wsum[waveInBlock] = contrib;
  __syncthreads();
  if (threadIdx.x == 0) {
    float s = 0.0f;
    const int nw = blockDim.x >> 5;
#pragma unroll
    for (int w = 0; w < 8; ++w)
      if (w < nw) s += wsum[w];
    block_partials[blockIdx.x] = s;
  }
}

// ---------------------------------------------------------------------------
// Kernel 3: single-block deterministic tree reduction of block partials.
// ---------------------------------------------------------------------------
__global__ void final_reduce_kernel(const float* __restrict__ partials, int nb,
                                    float* __restrict__ out) {
  __shared__ float s[256];
  float acc = 0.0f;
  for (int i = threadIdx.x; i < nb; i += 256) acc += partials[i];
  s[threadIdx.x] = acc;
  __syncthreads();
  for (int off = 128; off > 0; off >>= 1) {
    if (threadIdx.x < off) s[threadIdx.x] += s[threadIdx.x + off];
    __syncthreads();
  }
  if (threadIdx.x == 0) out[0] = s[0];
}

// ---------------------------------------------------------------------------
extern "C" void kernel_launch(void* const* d_in, const int* in_sizes, int n_in,
                              void* d_out, int out_size, void* d_ws,
                              size_t ws_size, hipStream_t stream) {
  const float* x = (const float*)d_in[0];
  const int* trip = (const int*)d_in[1];
  float* out = (float*)d_out;

  const int nrows = in_sizes[0] / DIMS;   // 16384
  const int T = in_sizes[1] / 3;          // 524288

  float* sq = (float*)d_ws;               // nrows floats
  float* block_partials = sq + nrows;     // one float per block of kernel 2

  // Kernel 1: 8 waves/block -> 8 rows/block
  {
    int blocks = (nrows + 7) / 8;
    sqnorm_kernel<<<blocks, 256, 0, stream>>>(x, sq, nrows);
  }

  // Kernel 2: 16 triplets per wave, 8 waves (128 triplets) per block
  int waves = (T + 15) / 16;
  int blocks2 = (waves + 7) / 8;
  triplet_wmma_kernel<<<blocks2, 256, 0, stream>>>(x, trip, sq, block_partials,
                                                   T);

  // Kernel 3: deterministic final sum
  final_reduce_kernel<<<1, 256, 0, stream>>>(block_partials, blocks2, out);
}